// CTE_37512244364037
// MI455X (gfx1250) — compile-verified
//
#include <hip/hip_runtime.h>
#include <hip/hip_bf16.h>
#include <math.h>

#define M_FERNS 8
#define K_BITS  10
#define T_WORDS 4
#define D_OUTC  16
#define NCLS    10
#define N_IMG   32
#define C_IN    3
#define H_IMG   64
#define W_IMG   64
#define PH      (H_IMG/2)
#define PW      (W_IMG/2)
#define FLAT_K  (D_OUTC*PH*PW)   /* 16384 */

typedef __attribute__((ext_vector_type(2))) float v2f;
typedef __attribute__((ext_vector_type(4))) float v4f;
typedef __attribute__((ext_vector_type(8))) float v8f;

// ---------------------------------------------------------------------------
// Kernel 1: fern soft-bit features, T=4 active-word table accumulation,
// fused 2x2 average pool. One thread per pooled (n, ph, pw) location.
// x (1.5MB) and table (512KB) are L2-resident; fern params staged in LDS.
// ---------------------------------------------------------------------------
__global__ __launch_bounds__(256) void fern_feat_kernel(
    const float* __restrict__ x,
    const int*   __restrict__ c1,  const int* __restrict__ c2,
    const int*   __restrict__ dy1, const int* __restrict__ dx1,
    const int*   __restrict__ dy2, const int* __restrict__ dx2,
    const float* __restrict__ thr,
    const float* __restrict__ table,
    float*       __restrict__ pooled)          // [N, D_OUT, PH, PW]
{
    __shared__ int   s_c1[M_FERNS*K_BITS], s_c2[M_FERNS*K_BITS];
    __shared__ int   s_y1[M_FERNS*K_BITS], s_x1[M_FERNS*K_BITS];
    __shared__ int   s_y2[M_FERNS*K_BITS], s_x2[M_FERNS*K_BITS];
    __shared__ float s_th[M_FERNS*K_BITS];

    const int tid = threadIdx.x;
    if (tid < M_FERNS*K_BITS) {
        s_c1[tid] = c1[tid];  s_c2[tid] = c2[tid];
        s_y1[tid] = dy1[tid]; s_x1[tid] = dx1[tid];
        s_y2[tid] = dy2[tid]; s_x2[tid] = dx2[tid];
        s_th[tid] = thr[tid];
    }
    __syncthreads();

    const int idx = blockIdx.x * blockDim.x + tid;   // [0, 32*32*32)
    const int n   = idx >> 10;
    const int rem = idx & 1023;
    const int ph  = rem >> 5, pw = rem & 31;

    float feat[D_OUTC];
    #pragma unroll
    for (int d = 0; d < D_OUTC; ++d) feat[d] = 0.f;

    #pragma unroll
    for (int py = 0; py < 2; ++py)
    #pragma unroll
    for (int px = 0; px < 2; ++px) {
        const int h = 2*ph + py, w = 2*pw + px;
        for (int m = 0; m < M_FERNS; ++m) {
            // --- K soft bits + hard word + fused stable top-2 ambiguity ---
            int   base = 0;
            int   i0 = 0, i1 = 0;
            float a0 = 1e30f, a1 = 1e30f, b0 = 0.f, b1 = 0.f;
            #pragma unroll
            for (int k = 0; k < K_BITS; ++k) {
                const int p  = m*K_BITS + k;
                const int y1 = h + s_y1[p], xx1 = w + s_x1[p];
                const int y2 = h + s_y2[p], xx2 = w + s_x2[p];
                // bottom/right zero padding (offsets are in [0, L-1])
                const float v1 = (y1 < H_IMG && xx1 < W_IMG)
                    ? x[((size_t)(n*C_IN + s_c1[p])*H_IMG + y1)*W_IMG + xx1] : 0.f;
                const float v2 = (y2 < H_IMG && xx2 < W_IMG)
                    ? x[((size_t)(n*C_IN + s_c2[p])*H_IMG + y2)*W_IMG + xx2] : 0.f;
                const float dlt = v1 - v2 - s_th[p];
                const float b   = 1.f / (1.f + __expf(-dlt));   // TEMP = 1
                base |= (dlt > 0.f) ? (1 << k) : 0;             // b > 0.5 <=> dlt > 0
                const float a = fabsf(b - 0.5f);
                // strict '<' keeps the earliest index on ties (matches top_k)
                if (a < a0)      { a1 = a0; b1 = b0; i1 = i0; a0 = a; b0 = b; i0 = k; }
                else if (a < a1) { a1 = a;  b1 = b;  i1 = k; }
            }
            const int p0 = 1 << i0, p1 = 1 << i1;
            const int bc = base & ~(p0 | p1);   // clear ambiguous bits
            const float* tb = table + ((size_t)(m << K_BITS)) * D_OUTC;
            // bits[t,a] = (t>>a)&1 ; a=0 -> (i0,b0), a=1 -> (i1,b1)
            #pragma unroll
            for (int t = 0; t < T_WORDS; ++t) {
                const int   word = bc + ((t & 1) ? p0 : 0) + ((t & 2) ? p1 : 0);
                const float wgt  = ((t & 1) ? b0 : 1.f - b0)
                                 * ((t & 2) ? b1 : 1.f - b1);
                const v4f* row = (const v4f*)(tb + (size_t)word * D_OUTC);
                #pragma unroll
                for (int q = 0; q < 4; ++q) {
                    const v4f r = row[q];               // global_load_b128
                    feat[4*q+0] += wgt * r.x;
                    feat[4*q+1] += wgt * r.y;
                    feat[4*q+2] += wgt * r.z;
                    feat[4*q+3] += wgt * r.w;
                }
            }
        }
    }
    // 2x2 average pool; pooled layout == reference flat layout [n][d][ph][pw]
    #pragma unroll
    for (int d = 0; d < D_OUTC; ++d)
        pooled[((size_t)n*D_OUTC + d)*(PH*PW) + ph*PW + pw] = 0.25f * feat[d];
}

// ---------------------------------------------------------------------------
// Kernel 2: prediction head on the matrix pipe.
// out[32,10] = flat[32,16384] @ w_pred^T[16384,10] + b_pred
// V_WMMA_F32_16X16X4_F32, full f32 precision. One wave per 16-row tile.
// A 16x4 f32: lanes 0-15 carry K={0,1} (2 VGPRs), lanes 16-31 K={2,3}.
// B 4x16 f32: same K split per lane half, N = lane&15.
// D 16x16 f32: VGPR i -> M=i (lanes 0-15) / M=8+i (lanes 16-31), N = lane&15.
// ---------------------------------------------------------------------------
__global__ __launch_bounds__(32) void head_gemm_wmma(
    const float* __restrict__ flat,     // [32, 16384]
    const float* __restrict__ w_pred,   // [10, 16384]
    const float* __restrict__ b_pred,   // [10]
    float*       __restrict__ out)      // [32, 10]
{
    const int lane = threadIdx.x;               // 0..31, one wave32
    const int r0   = blockIdx.x * 16;           // row tile base
    const int mrow = lane & 15;
    const int koff = (lane >> 4) << 1;          // 0 or 2
    const int ncol = lane & 15;

    const float* arow = flat + (size_t)(r0 + mrow) * FLAT_K + koff;
    // pad N 10->16: clamp pointer in-bounds, zero via mask (no divergence,
    // EXEC is all-ones at every WMMA as required)
    const int    nc   = (ncol < NCLS) ? ncol : (NCLS - 1);
    const float* brow = w_pred + (size_t)nc * FLAT_K + koff;
    const float  bmask = (ncol < NCLS) ? 1.f : 0.f;

    v8f acc = {0.f, 0.f, 0.f, 0.f, 0.f, 0.f, 0.f, 0.f};
    #pragma unroll 8
    for (int k = 0; k < FLAT_K; k += 4) {
        v2f a, b;
        a.x = arow[k];            a.y = arow[k + 1];        // b64 load
        b.x = bmask * brow[k];    b.y = bmask * brow[k + 1];
        acc = __builtin_amdgcn_wmma_f32_16x16x4_f32(
            /*neg_a=*/false, a, /*neg_b=*/false, b,
            /*c_mod=*/(short)0, acc, /*reuse_a=*/false, /*reuse_b=*/false);
    }

    if (ncol < NCLS) {
        const float bias  = b_pred[ncol];
        const int   rbase = r0 + ((lane >> 4) << 3);        // +0 or +8
        #pragma unroll
        for (int i = 0; i < 8; ++i)
            out[(size_t)(rbase + i) * NCLS + ncol] = acc[i] + bias;
    }
}

// ---------------------------------------------------------------------------
extern "C" void kernel_launch(void* const* d_in, const int* in_sizes, int n_in,
                              void* d_out, int out_size, void* d_ws, size_t ws_size,
                              hipStream_t stream) {
    const float* x      = (const float*)d_in[0];
    const int*   c1     = (const int*)  d_in[1];
    const int*   c2     = (const int*)  d_in[2];
    const int*   dy1    = (const int*)  d_in[3];
    const int*   dx1    = (const int*)  d_in[4];
    const int*   dy2    = (const int*)  d_in[5];
    const int*   dx2    = (const int*)  d_in[6];
    const float* thr    = (const float*)d_in[7];
    const float* table  = (const float*)d_in[8];
    const float* w_pred = (const float*)d_in[9];
    const float* b_pred = (const float*)d_in[10];
    float*       out    = (float*)d_out;
    float*       pooled = (float*)d_ws;         // 32*16384 f32 = 2 MB scratch

    const int total = N_IMG * PH * PW;          // 32768 threads
    fern_feat_kernel<<<dim3((total + 255) / 256), dim3(256), 0, stream>>>(
        x, c1, c2, dy1, dx1, dy2, dx2, thr, table, pooled);

    head_gemm_wmma<<<dim3(2), dim3(32), 0, stream>>>(pooled, w_pred, b_pred, out);
}